// Attention_82755429859383
// MI455X (gfx1250) — compile-verified
//
#include <hip/hip_runtime.h>
#include <hip/hip_bf16.h>
#include <math.h>

// ---------------- problem constants (from reference) ----------------
constexpr int B   = 16;
constexpr int L   = 4;
constexpr int D   = 2048;
constexpr int H   = 16;
constexpr int HD  = 128;
constexpr int KV  = 4096;
constexpr int KVL = KV + L;          // 4100
constexpr int MROWS = B * L;         // 64
constexpr float SCALE = 0.08838834764831845f;  // 128^-0.5
constexpr float NEG_BIG = -1.0e30f;

constexpr int AWAVES = 8;            // waves per attention block
constexpr int HDP = HD + 4;          // padded row stride (bank-conflict-free)

typedef __attribute__((ext_vector_type(2))) float v2f;
typedef __attribute__((ext_vector_type(8))) float v8f;
typedef int v4i __attribute__((vector_size(16)));

#if __has_builtin(__builtin_amdgcn_global_load_async_to_lds_b128) && \
    __has_builtin(__builtin_amdgcn_s_wait_asynccnt)
#define USE_ASYNC_LDS 1
#else
#define USE_ASYNC_LDS 0
#endif

#define TO_GLB(p) ((__attribute__((address_space(1))) v4i*)(p))
#define TO_LDS(p) ((__attribute__((address_space(3))) v4i*)(p))

__device__ __forceinline__ v8f wmma_f32(v2f a, v2f b, v8f c) {
    // fp32 WMMA: D(16x16) = A(16x4) * B(4x16) + C  -> v_wmma_f32_16x16x4_f32
    return __builtin_amdgcn_wmma_f32_16x16x4_f32(
        /*neg_a=*/false, a, /*neg_b=*/false, b,
        /*c_mod=*/(short)0, c, /*reuse_a=*/false, /*reuse_b=*/false);
}

// ---------------- kernel 1: cache copy into concatenated outputs -------------
__global__ void copy_cache_kernel(const float4* __restrict__ kc,
                                  const float4* __restrict__ vc,
                                  float4* __restrict__ keys,
                                  float4* __restrict__ vals) {
    const size_t per_bh_src = (size_t)KV * HD / 4;    // 131072 float4 per (b,h)
    const size_t per_bh_dst = (size_t)KVL * HD / 4;   // 131200 float4 per (b,h)
    size_t idx = (size_t)blockIdx.x * blockDim.x + threadIdx.x;
    const size_t total = (size_t)B * H * per_bh_src;
    if (idx >= total) return;
    size_t bh  = idx / per_bh_src;
    size_t rem = idx - bh * per_bh_src;
    size_t dst = bh * per_bh_dst + rem;
    keys[dst] = kc[idx];
    vals[dst] = vc[idx];
}

// ---------------- kernel 2: O = X @ W^T  (M=64, N=2048, K=2048, fp32 WMMA) ---
__global__ void gemm_xwT_kernel(const float* __restrict__ X,
                                const float* __restrict__ W,
                                float* __restrict__ O) {
    const int NT = D / 16;                      // 128 n-tiles
    const int MT = MROWS / 16;                  // 4 m-tiles
    int wid  = threadIdx.x >> 5;
    int lane = threadIdx.x & 31;
    int tile = blockIdx.x * (blockDim.x >> 5) + wid;
    if (tile >= MT * NT) return;                // wave-uniform
    int mt = tile / NT;
    int nt = tile - mt * NT;
    int half = lane >> 4;                       // 0: K 0/1, 1: K 2/3
    int lq   = lane & 15;

    const float* arow = X + (size_t)(mt * 16 + lq) * D;   // A[m=lq][k]
    const float* brow = W + (size_t)(nt * 16 + lq) * D;   // B[k][n=lq] = W[n][k]

    v8f c = {};
    for (int k0 = 0; k0 < D; k0 += 4) {
        v2f a, b;
        a.x = arow[k0 + 2 * half];
        a.y = arow[k0 + 2 * half + 1];
        b.x = brow[k0 + 2 * half];
        b.y = brow[k0 + 2 * half + 1];
        c = wmma_f32(a, b, c);
    }
    // C layout: element i -> row m = i + 8*half, col n = lq
#pragma unroll
    for (int i = 0; i < 8; ++i) {
        int m = i + 8 * half;
        O[(size_t)(mt * 16 + m) * D + nt * 16 + lq] = c[i];
    }
}

// ---------------- kernel 3: RoPE q (in place) + append roped k, raw v --------
__global__ void rope_append_kernel(float* __restrict__ q_ws,
                                   const float* __restrict__ k_ws,
                                   const float* __restrict__ v_ws,
                                   float* __restrict__ keys,
                                   float* __restrict__ vals) {
    int idx = blockIdx.x * blockDim.x + threadIdx.x;
    const int total = B * L * H * (HD / 2);     // 65536
    if (idx >= total) return;
    int p = idx & 63;         int t = idx >> 6;
    int h = t & (H - 1);      t >>= 4;
    int l = t & (L - 1);      int b = t >> 2;

    const float lnbase = 9.210340371976184f;    // ln(10000)
    float inv_freq = __expf(-((float)(2 * p) / (float)HD) * lnbase);
    float pos = (float)(KV + l);
    float fr = pos * inv_freq;
    float cs = cosf(fr), sn = sinf(fr);

    size_t row = (size_t)(b * L + l) * D + h * HD + 2 * p;
    float q1 = q_ws[row], q2 = q_ws[row + 1];
    q_ws[row]     = q1 * cs - q2 * sn;
    q_ws[row + 1] = q1 * sn + q2 * cs;

    float k1 = k_ws[row], k2 = k_ws[row + 1];
    size_t kvrow = (((size_t)(b * H + h) * KVL) + KV + l) * HD + 2 * p;
    keys[kvrow]     = k1 * cs - k2 * sn;
    keys[kvrow + 1] = k1 * sn + k2 * cs;

    vals[kvrow]     = v_ws[row];
    vals[kvrow + 1] = v_ws[row + 1];
}

// ---------------- kernel 4: flash attention per (b,h), fp32 WMMA -------------
// 8 waves/block; each wave streams 16-row KV tiles; K tile staged to LDS via
// the gfx1250 async-to-LDS path (ASYNCcnt) when available.
__global__ void __launch_bounds__(32 * AWAVES)
attn_kernel(const float* __restrict__ q_ws,     // (B,L,D) roped
            const float* __restrict__ keys,     // (B,H,KVL,HD)
            const float* __restrict__ vals,     // (B,H,KVL,HD)
            const float* __restrict__ mask,     // (1,1,L,KVL)
            float* __restrict__ attn_out) {     // (B,L,D)
    __shared__ float q_s[16][HDP];                   // Q padded to 16 rows
    __shared__ float kbuf[AWAVES][16][HDP];          // per-wave K tile
    __shared__ float p_s[AWAVES][16][17];            // per-wave P staging
    __shared__ float red_m[AWAVES][L];
    __shared__ float red_l[AWAVES][L];
    __shared__ float red_acc[AWAVES][L][HD];

    const int bh = blockIdx.x;
    const int b  = bh / H;
    const int h  = bh - b * H;
    const int tid  = threadIdx.x;
    const int wid  = tid >> 5;
    const int lane = tid & 31;
    const int half = lane >> 4;
    const int lq   = lane & 15;
    const int nthr = 32 * AWAVES;

    // stage Q (rows >= L zeroed) and zero P staging
    for (int i = tid; i < 16 * HDP; i += nthr) {
        int r = i / HDP, c = i - r * HDP;
        q_s[r][c] = (r < L && c < HD)
                        ? q_ws[(size_t)(b * L + r) * D + h * HD + c] : 0.0f;
    }
    for (int i = tid; i < AWAVES * 16 * 17; i += nthr)
        ((float*)p_s)[i] = 0.0f;
    __syncthreads();

    float m_r[L], l_r[L];
#pragma unroll
    for (int r = 0; r < L; ++r) { m_r[r] = NEG_BIG; l_r[r] = 0.0f; }
    union { v8f v; float f[8]; } acc[8];        // 8 dim-groups of 16
#pragma unroll
    for (int g = 0; g < 8; ++g) acc[g].v = (v8f){};

    const int NTILES = (KVL + 15) / 16;         // 257
    const size_t base = (size_t)bh * KVL;

    for (int t = wid; t < NTILES; t += AWAVES) {
        const int kv0 = t * 16;
        const int kvn = kv0 + lq;
        const bool kok = kvn < KVL;

#if USE_ASYNC_LDS
        // ---- async-stage the 16x128 K tile: 16 x b128 ops, 512B each --------
#pragma unroll
        for (int it = 0; it < 16; ++it) {
            int chunk = it * 32 + lane;          // 0..511 : 16B chunks
            int row   = chunk >> 5;              // 0..15
            int col   = (chunk & 31) << 2;       // float column
            int grow  = kv0 + row;
            if (grow > KVL - 1) grow = KVL - 1;  // clamp; masked later
            __builtin_amdgcn_global_load_async_to_lds_b128(
                TO_GLB(keys + (base + grow) * HD + col),
                TO_LDS(&kbuf[wid][row][col]), 0, 0);
        }
        __builtin_amdgcn_s_wait_asynccnt(0);
        asm volatile("" ::: "memory");
#else
        // ---- fallback: direct-stage K tile through VGPRs --------------------
        {
            const float4* src;
#pragma unroll
            for (int it = 0; it < 16; ++it) {
                int chunk = it * 32 + lane;
                int row   = chunk >> 5;
                int col   = (chunk & 31) << 2;
                int grow  = kv0 + row;
                if (grow > KVL - 1) grow = KVL - 1;
                src = (const float4*)(keys + (base + grow) * HD + col);
                *(float4*)&kbuf[wid][row][col] = *src;
            }
        }
        __threadfence_block();
#endif

        // prefetch next tile's key rows for this wave
        if (kv0 + 16 * AWAVES + lq < KVL)
            __builtin_prefetch(keys + (base + kv0 + 16 * AWAVES + lq) * HD, 0, 0);

        // ---- S tile = Q(16x128) x K^T chunk(128x16), K-loop in steps of 4 ---
        v8f S = {};
        for (int k0 = 0; k0 < HD; k0 += 4) {
            v2f a, bb;
            a.x  = q_s[lq][k0 + 2 * half];
            a.y  = q_s[lq][k0 + 2 * half + 1];
            bb.x = kbuf[wid][lq][k0 + 2 * half];     // B[k][n=lq] = K[kv0+lq][k]
            bb.y = kbuf[wid][lq][k0 + 2 * half + 1];
            S = wmma_f32(a, bb, S);
        }

        // ---- online softmax over the L=4 real rows --------------------------
        float fcorr[L];
#pragma unroll
        for (int r = 0; r < L; ++r) {
            float s = S[r] * SCALE;
            if (kok) s += mask[r * KVL + kvn];
            if (!kok || half != 0) s = NEG_BIG;
            float mx = s;
            mx = fmaxf(mx, __shfl_xor(mx, 8, 16));
            mx = fmaxf(mx, __shfl_xor(mx, 4, 16));
            mx = fmaxf(mx, __shfl_xor(mx, 2, 16));
            mx = fmaxf(mx, __shfl_xor(mx, 1, 16));
            mx = __shfl(mx, 0, 32);
            float mnew = fmaxf(m_r[r], mx);
            float pv = (kok && half == 0) ? __expf(s - mnew) : 0.0f;
            if (half == 0) p_s[wid][r][lq] = pv;
            float ls = pv;
            ls += __shfl_xor(ls, 8, 16);
            ls += __shfl_xor(ls, 4, 16);
            ls += __shfl_xor(ls, 2, 16);
            ls += __shfl_xor(ls, 1, 16);
            ls = __shfl(ls, 0, 32);
            fcorr[r] = __expf(m_r[r] - mnew);
            l_r[r] = l_r[r] * fcorr[r] + ls;
            m_r[r] = mnew;
        }
#pragma unroll
        for (int g = 0; g < 8; ++g)
#pragma unroll
            for (int r = 0; r < L; ++r) acc[g].f[r] *= fcorr[r];

        __threadfence_block();  // LDS store -> cross-lane load ordering in-wave

        // ---- acc += P(16x16) x V chunk(16x128) ------------------------------
#pragma unroll
        for (int kk = 0; kk < 4; ++kk) {
            v2f a;
            a.x = p_s[wid][lq][4 * kk + 2 * half];
            a.y = p_s[wid][lq][4 * kk + 2 * half + 1];
            int kr = kv0 + 4 * kk + 2 * half;
#pragma unroll
            for (int g = 0; g < 8; ++g) {
                v2f bb;
                bb.x = (kr     < KVL) ? vals[(base + kr)     * HD + g * 16 + lq] : 0.0f;
                bb.y = (kr + 1 < KVL) ? vals[(base + kr + 1) * HD + g * 16 + lq] : 0.0f;
                acc[g].v = wmma_f32(a, bb, acc[g].v);
            }
        }
    }

    // ---- per-wave partials to LDS ------------------------------------------
    if (lane == 0) {
#pragma unroll
        for (int r = 0; r < L; ++r) { red_m[wid][r] = m_r[r]; red_l[wid][r] = l_r[r]; }
    }
    if (half == 0) {
#pragma unroll
        for (int g = 0; g < 8; ++g)
#pragma unroll
            for (int r = 0; r < L; ++r)
                red_acc[wid][r][g * 16 + lq] = acc[g].f[r];
    }
    __syncthreads();

    // ---- merge AWAVES waves, normalize, write (B,L,H*HD) --------------------
    for (int i = tid; i < L * HD; i += nthr) {
        int r = i / HD, dc = i - r * HD;
        float M = NEG_BIG;
#pragma unroll
        for (int w = 0; w < AWAVES; ++w) M = fmaxf(M, red_m[w][r]);
        float Ls = 0.0f, o = 0.0f;
#pragma unroll
        for (int w = 0; w < AWAVES; ++w) {
            float e = __expf(red_m[w][r] - M);
            Ls += red_l[w][r] * e;
            o  += red_acc[w][r][dc] * e;
        }
        attn_out[(size_t)(b * L + r) * D + h * HD + dc] = o / Ls;
    }
}

// ---------------- host side --------------------------------------------------
extern "C" void kernel_launch(void* const* d_in, const int* in_sizes, int n_in,
                              void* d_out, int out_size, void* d_ws, size_t ws_size,
                              hipStream_t stream) {
    const float* x    = (const float*)d_in[0];
    const float* mask = (const float*)d_in[1];
    const float* kc   = (const float*)d_in[2];
    const float* vc   = (const float*)d_in[3];
    const float* q_w  = (const float*)d_in[4];
    const float* k_w  = (const float*)d_in[5];
    const float* v_w  = (const float*)d_in[6];
    const float* o_w  = (const float*)d_in[7];

    float* out  = (float*)d_out;                       // (B,L,D)
    float* keys = out + (size_t)MROWS * D;             // (B,H,KVL,HD)
    float* vals = keys + (size_t)B * H * KVL * HD;     // (B,H,KVL,HD)

    float* q_ws = (float*)d_ws;                        // (B,L,D)
    float* k_ws = q_ws + (size_t)MROWS * D;
    float* v_ws = k_ws + (size_t)MROWS * D;
    float* a_ws = v_ws + (size_t)MROWS * D;            // attention out pre-o_w

    // 1) stream caches into concatenated outputs
    {
        size_t total4 = (size_t)B * H * KV * HD / 4;   // 33,554,432 float4
        int blocks = (int)((total4 + 255) / 256);
        copy_cache_kernel<<<blocks, 256, 0, stream>>>(
            (const float4*)kc, (const float4*)vc, (float4*)keys, (float4*)vals);
    }
    // 2) Q/K/V projections (fp32 WMMA)
    gemm_xwT_kernel<<<64, 256, 0, stream>>>(x, q_w, q_ws);
    gemm_xwT_kernel<<<64, 256, 0, stream>>>(x, k_w, k_ws);
    gemm_xwT_kernel<<<64, 256, 0, stream>>>(x, v_w, v_ws);
    // 3) RoPE + append new k/v rows
    rope_append_kernel<<<256, 256, 0, stream>>>(q_ws, k_ws, v_ws, keys, vals);
    // 4) attention (one block per (b,h), 8 waves)
    attn_kernel<<<B * H, 32 * AWAVES, 0, stream>>>(q_ws, keys, vals, mask, a_ws);
    // 5) output projection
    gemm_xwT_kernel<<<64, 256, 0, stream>>>(a_ws, o_w, out);
}